// AuditorySpectrogram_59854664237430
// MI455X (gfx1250) — compile-verified
//
#include <hip/hip_runtime.h>
#include <hip/hip_bf16.h>
#include <math.h>

// ---------------------------------------------------------------------------
// Auditory spectrogram for MI455X (gfx1250, wave32).
//   N = 524288 = 8 * 16^4  -> Stockham mixed-radix FFT, radix-16 stages done
//   as complex 16x16 DFT matmuls on V_WMMA_F32_16X16X4_F32.
//   Twiddles use hardware sin/cos (__sincosf) after exact integer range
//   reduction, keeping the stage WMMA/bandwidth-bound instead of VALU-bound.
// ---------------------------------------------------------------------------

#define NFFT 524288
#define CCH  129
#define DSR  80
#define KIIR 2560           // alpha^2560 ~ 2e-9: below fp32 noise
#define JBLK (KIIR / DSR)   // 32

typedef __attribute__((ext_vector_type(2))) float v2f;
typedef __attribute__((ext_vector_type(8))) float v8f;

__device__ __forceinline__ v8f wmma4(v2f a, v2f b, v8f c) {
  // D(16x16) += A(16x4) * B(4x16), fp32. 8-arg form:
  // (neg_a, A, neg_b, B, c_mod, C, reuse_a, reuse_b). NEG on A/B illegal for
  // F32 type per ISA (NEG = CNeg only), so always false.
  return __builtin_amdgcn_wmma_f32_16x16x4_f32(false, a, false, b, (short)0, c,
                                               false, false);
}

// ---------------------------------------------------------------------------
// Stockham stage, radix 16 via WMMA. One wave = one tile of 16 butterflies.
//   in[b + (N/16)*r]  --twiddle W_{16L}^{r*l}-->  U(16x16)
//   V = F16 * U  (complex, F16[j][r] = e^{i*dir*2pi*j*r/16})
//   out[l + L*j + 16*L*m'] = V[j][b],  b = l + L*m'
// ---------------------------------------------------------------------------
__global__ __launch_bounds__(256) void fft_stage16(
    const float* __restrict__ inRe, const float* __restrict__ inIm,
    float* __restrict__ outRe, float* __restrict__ outIm,
    long slotStride, int L, float dir)
{
  const int lane = threadIdx.x & 31;
  const int wave = threadIdx.x >> 5;
  const long soff = (long)blockIdx.y * slotStride;
  const float* iR = inRe + soff;
  const float* iI = inIm + soff;
  float* oR = outRe + soff;
  float* oI = outIm + soff;

  const int col = lane & 15;       // tile column = butterfly within tile
  const int hh  = lane >> 4;       // which half of the wave (K pairing)
  const int b   = (blockIdx.x * 8 + wave) * 16 + col;
  const int lsh = 31 - __builtin_clz(L);
  const int l   = b & (L - 1);
  const int mp  = b >> lsh;
  const int step = NFFT / 16;
  const int LR   = L * 16;
  const float TWO_PI = 6.28318530717958647692f;

  // DFT-16 matrix fragments (A operand). Lane<16 holds K={4kc,4kc+1},
  // lane>=16 holds K={4kc+2,4kc+3} per the 16x4 f32 A layout.
  // Angles are exact multiples of 2pi/16 in [0,2pi): HW sin/cos is plenty.
  v2f fre[4], fim[4], fimn[4];
#pragma unroll
  for (int kc = 0; kc < 4; ++kc) {
    const int r0 = 4 * kc + 2 * hh;
    const int r1 = r0 + 1;
    float s0, c0, s1, c1;
    __sincosf(dir * TWO_PI * (float)((col * r0) & 15) * (1.0f / 16.0f), &s0, &c0);
    __sincosf(dir * TWO_PI * (float)((col * r1) & 15) * (1.0f / 16.0f), &s1, &c1);
    fre[kc].x = c0;  fre[kc].y = c1;
    fim[kc].x = s0;  fim[kc].y = s1;
    fimn[kc].x = -s0; fimn[kc].y = -s1;
  }

  // Load + stage twiddle (B operand, same lane->K pairing as A).
  // (r*l) mod (16L) keeps the argument in (-2pi, 2pi) exactly.
  v2f ure[4], uim[4];
  const float asc = dir * TWO_PI / (float)LR;
#pragma unroll
  for (int kc = 0; kc < 4; ++kc) {
    const int r0 = 4 * kc + 2 * hh;
    const int r1 = r0 + 1;
    const float x0r = iR[b + step * r0], x0i = iI[b + step * r0];
    const float x1r = iR[b + step * r1], x1i = iI[b + step * r1];
    float t0s, t0c, t1s, t1c;
    __sincosf(asc * (float)((r0 * l) & (LR - 1)), &t0s, &t0c);
    __sincosf(asc * (float)((r1 * l) & (LR - 1)), &t1s, &t1c);
    ure[kc].x = t0c * x0r - t0s * x0i;
    uim[kc].x = t0c * x0i + t0s * x0r;
    ure[kc].y = t1c * x1r - t1s * x1i;
    uim[kc].y = t1c * x1i + t1s * x1r;
  }

  // Complex matmul: Vre = Fre*Ure - Fim*Uim ; Vim = Fre*Uim + Fim*Ure.
  v8f vre = {0.f, 0.f, 0.f, 0.f, 0.f, 0.f, 0.f, 0.f};
  v8f vim = {0.f, 0.f, 0.f, 0.f, 0.f, 0.f, 0.f, 0.f};
#pragma unroll
  for (int kc = 0; kc < 4; ++kc) vre = wmma4(fre[kc], ure[kc], vre);
#pragma unroll
  for (int kc = 0; kc < 4; ++kc) vre = wmma4(fimn[kc], uim[kc], vre);
#pragma unroll
  for (int kc = 0; kc < 4; ++kc) vim = wmma4(fre[kc], uim[kc], vim);
#pragma unroll
  for (int kc = 0; kc < 4; ++kc) vim = wmma4(fim[kc], ure[kc], vim);

  // D layout: VGPR v holds rows j=v (lanes<16) and j=v+8 (lanes>=16).
  const long obase = (long)l + 16L * (long)L * (long)mp;
#pragma unroll
  for (int v = 0; v < 8; ++v) {
    const int j = v + 8 * hh;
    oR[obase + (long)L * j] = vre[v];
    oI[obase + (long)L * j] = vim[v];
  }
}

// Stockham stage, radix 8 (first stage, L=1 -> trivial twiddles). Plain VALU.
__global__ __launch_bounds__(256) void fft_stage8(
    const float* __restrict__ inRe, const float* __restrict__ inIm,
    float* __restrict__ outRe, float* __restrict__ outIm,
    long slotStride, float dir)
{
  const long soff = (long)blockIdx.y * slotStride;
  const float* iR = inRe + soff;
  const float* iI = inIm + soff;
  float* oR = outRe + soff;
  float* oI = outIm + soff;
  const int bfly = blockIdx.x * 256 + threadIdx.x;   // [0, N/8)
  const int step = NFFT / 8;

  float xr[8], xi[8], wr[8], wi[8];
#pragma unroll
  for (int r = 0; r < 8; ++r) {
    xr[r] = iR[bfly + step * r];
    xi[r] = iI[bfly + step * r];
  }
#pragma unroll
  for (int j = 0; j < 8; ++j) {
    float s, c;
    __sincosf(dir * 0.78539816339744830962f * (float)j, &s, &c);
    wr[j] = c; wi[j] = s;
  }
#pragma unroll
  for (int j = 0; j < 8; ++j) {
    float sr = 0.f, si = 0.f;
#pragma unroll
    for (int r = 0; r < 8; ++r) {
      const int p = (j * r) & 7;
      sr = fmaf(wr[p], xr[r], fmaf(-wi[p], xi[r], sr));
      si = fmaf(wr[p], xi[r], fmaf( wi[p], xr[r], si));
    }
    oR[8 * bfly + j] = sr;
    oI[8 * bfly + j] = si;
  }
}

__global__ __launch_bounds__(256) void init_fwd(const float* __restrict__ x,
                                                float* __restrict__ are,
                                                float* __restrict__ aim)
{
  const int i = blockIdx.x * 256 + threadIdx.x;
  are[i] = x[i];
  aim[i] = 0.f;
}

__global__ __launch_bounds__(256) void wtab_kernel(const float* __restrict__ aleak,
                                                   float* __restrict__ wtab)
{
  const int k = blockIdx.x * 256 + threadIdx.x;
  if (k < KIIR) wtab[k] = powf(aleak[0], (float)k);
}

// spectrum[g] = (1/N) * Hs[c0+g] * X   (feeds the inverse transform)
// Hs is a 541MB single-use stream: NT loads + prefetch keep it out of the L2
// working set reserved for the FFT ping-pong planes.
__global__ __launch_bounds__(256) void specmul_kernel(
    const float* __restrict__ Hr, const float* __restrict__ Hi,
    const float* __restrict__ Xre, const float* __restrict__ Xim,
    float* __restrict__ outRe, float* __restrict__ outIm,
    long slotStride, int c0)
{
  const int g = blockIdx.y;
  const long i = (long)blockIdx.x * 256 + threadIdx.x;
  const long hoff = (long)(c0 + g) * NFFT + i;
  __builtin_prefetch(Hr + hoff + 8192, 0, 0);   // -> global_prefetch_b8
  __builtin_prefetch(Hi + hoff + 8192, 0, 0);
  const float hr = __builtin_nontemporal_load(Hr + hoff);
  const float hi = __builtin_nontemporal_load(Hi + hoff);
  const float xr = Xre[i], xi = Xim[i];
  const float inv = 1.0f / (float)NFFT;
  const long o = (long)g * slotStride + i;
  outRe[o] = (hr * xr - hi * xi) * inv;
  outIm[o] = (hr * xi + hi * xr) * inv;
}

// comp = |Re(coch)|^alpha * sign(Re(coch))
__global__ __launch_bounds__(256) void comp_kernel(
    const float* __restrict__ cochRe, float* __restrict__ compOut,
    long slotStride, const float* __restrict__ acomp, int c0)
{
  const int g = blockIdx.y;
  const long i = (long)blockIdx.x * 256 + threadIdx.x;
  const float v = cochRe[(long)g * slotStride + i];
  const float m = powf(fabsf(v), acomp[c0 + g]);
  compOut[(long)g * slotStride + i] = (v > 0.f) ? m : ((v < 0.f) ? -m : 0.f);
}

// lin = relu(k0*comp[c] + k1*comp[c+1]); top slot of batch uses carry plane.
__global__ __launch_bounds__(256) void lin_kernel(
    const float* __restrict__ compBase, float* __restrict__ linOut,
    long slotStride, const float* __restrict__ carry,
    const float* __restrict__ lik, int sz)
{
  const int g = blockIdx.y;
  const long i = (long)blockIdx.x * 256 + threadIdx.x;
  const float c  = compBase[(long)g * slotStride + i];
  const float cn = (g == sz - 1) ? carry[i]
                                 : compBase[(long)(g + 1) * slotStride + i];
  const float v = lik[0] * c + lik[1] * cn;
  linOut[(long)g * slotStride + i] = fmaxf(v, 0.f);
}

__global__ __launch_bounds__(256) void carry_copy(float* __restrict__ carry,
                                                  const float* __restrict__ comp0)
{
  const int i = blockIdx.x * 256 + threadIdx.x;
  carry[i] = comp0[i];
}

// Leaky integrator, phase 1: 80-tap partial sums on the downsample grid,
// extended JBLK blocks into the circular tail (handles FFT wrap-around).
__global__ __launch_bounds__(256) void iir_partial_kernel(
    const float* __restrict__ linBase, long slotStride,
    const float* __restrict__ wtab, float* __restrict__ S, int sz, int T)
{
  const long tid = (long)blockIdx.x * 256 + threadIdx.x;
  const int TE = T + JBLK;
  if (tid >= (long)sz * TE) return;
  const int g = (int)(tid / TE);
  const int u = (int)(tid % TE);
  const float* lin = linBase + (long)g * slotStride;
  int base = (DSR - 1) + DSR * (u - JBLK);
  if (u < JBLK) base += NFFT;          // circular wrap into the signal tail
  float s = 0.f;
#pragma unroll 8
  for (int k = 0; k < DSR; ++k) s = fmaf(wtab[k], lin[base - k], s);
  S[tid] = s;
}

// Phase 2: combine 32 partial blocks with weights alpha^(80j).
__global__ __launch_bounds__(256) void iir_combine_kernel(
    const float* __restrict__ S, const float* __restrict__ wtab,
    float* __restrict__ out, int c0, int sz, int T)
{
  const long tid = (long)blockIdx.x * 256 + threadIdx.x;
  if (tid >= (long)sz * T) return;
  const int g = (int)(tid / T);
  const int t = (int)(tid % T);
  const float* Sg = S + (long)g * (T + JBLK);
  float s = 0.f;
#pragma unroll
  for (int j = 0; j < JBLK; ++j) s = fmaf(wtab[DSR * j], Sg[t - j + JBLK], s);
  out[(long)(c0 + g) * T + t] = s;
}

// ---------------------------------------------------------------------------
extern "C" void kernel_launch(void* const* d_in, const int* in_sizes, int n_in,
                              void* d_out, int out_size, void* d_ws,
                              size_t ws_size, hipStream_t stream)
{
  (void)in_sizes; (void)n_in;
  const float* x     = (const float*)d_in[0];
  const float* Hr    = (const float*)d_in[1];
  const float* Hi    = (const float*)d_in[2];
  const float* acomp = (const float*)d_in[3];
  const float* aleak = (const float*)d_in[4];
  const float* lik   = (const float*)d_in[5];
  float* out = (float*)d_out;

  const int N = NFFT, C = CCH;
  const int T = out_size / C;           // 6553
  const int TE = T + JBLK;

  // Workspace layout (floats): X(2N) | carry(N) | wtab | Sbuf | batch slots
  float* wsf   = (float*)d_ws;
  float* Xre   = wsf;
  float* Xim   = Xre + N;
  float* carry = Xim + N;
  float* wtab  = carry + N;
  const int GMAX = 8;                   // keep batch ping-pong inside 192MB L2
  float* Sbuf  = wtab + KIIR;
  float* batch = Sbuf + (long)GMAX * TE;

  const size_t used = (size_t)(batch - wsf);
  const size_t wsfN = ws_size / sizeof(float);
  long avail = (long)wsfN - (long)used;
  int G = (avail > 0) ? (int)(avail / (4L * N)) : 0;
  if (G < 1) G = 1;
  if (G > GMAX) G = GMAX;
  const long SS = 4L * N;               // slot stride (4 planes per channel)

  float* A0r = batch;                   // slot-0 plane bases
  float* A0i = batch + (long)N;
  float* B0r = batch + 2L * N;
  float* B0i = batch + 3L * N;

  const dim3 blk(256, 1, 1);

  hipMemsetAsync(carry, 0, (size_t)N * sizeof(float), stream);
  wtab_kernel<<<dim3((KIIR + 255) / 256), blk, 0, stream>>>(aleak, wtab);

  // ---- forward FFT of x (dir = -1), result lands in Xre/Xim --------------
  init_fwd<<<dim3(N / 256), blk, 0, stream>>>(x, A0r, A0i);
  fft_stage8 <<<dim3(N / 8 / 256, 1), blk, 0, stream>>>(A0r, A0i, B0r, B0i, SS, -1.f);
  fft_stage16<<<dim3(256, 1), blk, 0, stream>>>(B0r, B0i, A0r, A0i, SS, 8,     -1.f);
  fft_stage16<<<dim3(256, 1), blk, 0, stream>>>(A0r, A0i, B0r, B0i, SS, 128,   -1.f);
  fft_stage16<<<dim3(256, 1), blk, 0, stream>>>(B0r, B0i, A0r, A0i, SS, 2048,  -1.f);
  fft_stage16<<<dim3(256, 1), blk, 0, stream>>>(A0r, A0i, Xre, Xim, SS, 32768, -1.f);

  // ---- per-channel batches, descending so the channel-diff carry works ---
  int c_end = C;
  while (c_end > 0) {
    const int sz = (c_end >= G) ? G : c_end;
    const int c0 = c_end - sz;
    const dim3 gN(N / 256, sz);
    const dim3 g8(N / 8 / 256, sz);
    const dim3 g16(256, sz);

    specmul_kernel<<<gN, blk, 0, stream>>>(Hr, Hi, Xre, Xim, A0r, A0i, SS, c0);

    // inverse FFT (dir = +1, 1/N already folded into specmul); ends in B.
    fft_stage8 <<<g8,  blk, 0, stream>>>(A0r, A0i, B0r, B0i, SS, 1.f);
    fft_stage16<<<g16, blk, 0, stream>>>(B0r, B0i, A0r, A0i, SS, 8,     1.f);
    fft_stage16<<<g16, blk, 0, stream>>>(A0r, A0i, B0r, B0i, SS, 128,   1.f);
    fft_stage16<<<g16, blk, 0, stream>>>(B0r, B0i, A0r, A0i, SS, 2048,  1.f);
    fft_stage16<<<g16, blk, 0, stream>>>(A0r, A0i, B0r, B0i, SS, 32768, 1.f);

    comp_kernel<<<gN, blk, 0, stream>>>(B0r, A0r, SS, acomp, c0);   // comp -> A_re
    lin_kernel <<<gN, blk, 0, stream>>>(A0r, A0i, SS, carry, lik, sz); // lin -> A_im
    carry_copy <<<dim3(N / 256), blk, 0, stream>>>(carry, A0r);     // carry = comp[c0]

    const long n1 = (long)sz * TE;
    iir_partial_kernel<<<dim3((unsigned)((n1 + 255) / 256)), blk, 0, stream>>>(
        A0i, SS, wtab, Sbuf, sz, T);
    const long n2 = (long)sz * T;
    iir_combine_kernel<<<dim3((unsigned)((n2 + 255) / 256)), blk, 0, stream>>>(
        Sbuf, wtab, out, c0, sz, T);

    c_end = c0;
  }
}